// RingPaddingLayer_11081015624173
// MI455X (gfx1250) — compile-verified
//
#include <hip/hip_runtime.h>
#include <stdint.h>
#include <stddef.h>

// Problem constants from the reference: x = (B, T, F) f32, wow = 16, F = 4.
#define B_      128
#define T_      16384
#define W_      16          // window width (wow)
#define TT_     512         // t-tile per workgroup
#define NTHREADS 256

struct alignas(16) f4 { float x, y, z, w; };

// ---- CDNA5 async Global<->LDS DMA (ASYNCcnt-tracked), via inline asm ----
// Input is consumed ~once -> non-temporal load; output is write-once streaming
// (512 MiB >> 192 MB L2) -> non-temporal store keeps L2 clean.
__device__ __forceinline__ void async_load_b128_nt(uint32_t lds_addr, const void* gaddr) {
    // each lane copies 16 bytes from the global address gaddr to the LDS address lds_addr
    asm volatile("global_load_async_to_lds_b128 %0, %1, off th:TH_LOAD_NT"
                 :: "v"(lds_addr), "v"(gaddr) : "memory");
}
__device__ __forceinline__ void async_store_b128_nt(void* gaddr, uint32_t lds_addr) {
    // each lane copies 16 bytes from the LDS address lds_addr to the global address gaddr
    asm volatile("global_store_async_from_lds_b128 %0, %1, off th:TH_STORE_NT"
                 :: "v"(gaddr), "v"(lds_addr) : "memory");
}
__device__ __forceinline__ void wait_asynccnt0() {
    asm volatile("s_wait_asynccnt 0" ::: "memory");
}

__global__ __launch_bounds__(NTHREADS)
void ring_window_kernel(const f4* __restrict__ x, f4* __restrict__ out) {
    // Staged input rows for this tile: x4[b][(t0-15 .. t0+TT_-1) mod T]
    __shared__ f4 tile[TT_ + W_];              // 528 * 16 B = 8448 B LDS

    const int b   = (int)(blockIdx.x >> 5);    // T_/TT_ == 32 tiles per b
    const int t0  = ((int)blockIdx.x & 31) * TT_;
    const int tid = (int)threadIdx.x;

    // Low 32 bits of the generic pointer to __shared__ == LDS byte offset.
    const uint32_t lds_base = (uint32_t)(uintptr_t)(&tile[0]);

    // ---- Phase 1: DMA the 527 needed input rows into LDS (coalesced 512B/wave) ----
    const int nrows = TT_ + W_ - 1;            // 527
    const f4* xb = x + (size_t)b * T_;
    for (int i = tid; i < nrows; i += NTHREADS) {
        int srct = t0 - (W_ - 1) + i;          // only underflow wrap possible
        if (srct < 0) srct += T_;
        async_load_b128_nt(lds_base + (uint32_t)i * 16u, (const void*)(xb + srct));
    }
    wait_asynccnt0();       // this wave's LDS writes have landed
    __syncthreads();        // all waves' LDS writes visible

    // ---- Phase 2: fan each staged row out 16x, LDS -> global, no VGPR round-trip ----
    // Flat tile output offset j = t_local*16 + w; lanes store 512 contiguous bytes
    // per wave-instruction. Per +NTHREADS step: dst advances 4096 B, LDS src 256 B.
    f4* dst = out + ((size_t)b * T_ + (size_t)t0) * W_ + tid;
    uint32_t lsrc = lds_base + (uint32_t)(((tid >> 4) + (tid & (W_ - 1))) * 16);
    #pragma unroll 8
    for (int k = 0; k < (TT_ * W_) / NTHREADS; ++k) {       // 32 stores/thread
        async_store_b128_nt((void*)dst, lsrc);
        dst  += NTHREADS;                                   // +4096 bytes
        lsrc += (NTHREADS / W_) * 16u;                      // +256 bytes
    }
    // s_endpgm performs an implicit wait-idle, draining outstanding async stores.
}

extern "C" void kernel_launch(void* const* d_in, const int* in_sizes, int n_in,
                              void* d_out, int out_size, void* d_ws, size_t ws_size,
                              hipStream_t stream) {
    (void)in_sizes; (void)n_in; (void)out_size; (void)d_ws; (void)ws_size;
    const f4* x  = (const f4*)d_in[0];   // d_in[1] is the scalar wow=16 (compile-time here)
    f4* out      = (f4*)d_out;
    dim3 grid(B_ * (T_ / TT_));          // 4096 blocks
    dim3 block(NTHREADS);
    hipLaunchKernelGGL(ring_window_kernel, grid, block, 0, stream, x, out);
}